// MultiHeadAttention_66056597013130
// MI455X (gfx1250) — compile-verified
//
#include <hip/hip_runtime.h>
#include <stdint.h>

typedef __attribute__((ext_vector_type(2))) float v2f;
typedef __attribute__((ext_vector_type(8))) float v8f;

#define DMODEL 1024
#define NHEADS 16
#define DKH    64
#define SEQ    2048
#define NBATCH 2

#define BM 64
#define BN 64
#define BK 16
#define ASTR 17               // LDS row stride (floats) for NT tiles
#define TILE_F (BM * ASTR)    // floats per staged tile (A or B)
#define BUF_F  (2 * TILE_F)   // A+B tiles per pipeline buffer
#define PSTR 68               // LDS row stride for P tile in PV kernel
#define VSTR 80               // LDS row stride for V tile in PV kernel

// D = A(16x4) * B(4x16) + C, all fp32, exact.
__device__ __forceinline__ v8f wmma_f32(v2f a, v2f b, v8f c) {
  return __builtin_amdgcn_wmma_f32_16x16x4_f32(false, a, false, b, (short)0, c,
                                               false, false);
}

// Async copy of 16 bytes (4 floats) per lane: global -> LDS, ASYNCcnt-tracked.
// lds_byte is the lane's 32-bit LDS byte offset (low 32 bits of generic ptr).
__device__ __forceinline__ void async_cp16(const float* g, uint32_t lds_byte) {
  asm volatile("global_load_async_to_lds_b128 %0, %1, off"
               :: "v"(lds_byte), "v"(g)
               : "memory");
}
#define WAIT_ASYNC(n) asm volatile("s_wait_asynccnt " #n ::: "memory")

// 64x64 output tile of C = A * B^T, A:[M,kdim] rows at A, B:[N,kdim] rows at B.
// 256 threads = 8 waves; wave w -> rows [16*(w%4),+16), cols [32*(w/4),+32).
// lds: 2 * BUF_F floats (double-buffered A+B tiles).
__device__ __forceinline__ void gemm_nt_tile(const float* __restrict__ A, int lda,
                                             const float* __restrict__ B, int ldb,
                                             int kdim, float* lds, v8f acc[2]) {
  const int t    = threadIdx.x;
  const int lane = t & 31;
  const int w    = t >> 5;
  const int wr   = w & 3;
  const int wc   = w >> 2;
  const int ml   = lane & 15;
  const int ksl  = (lane >> 4) << 1;  // 0 (lanes 0-15) or 2 (lanes 16-31)

  const int lr = t >> 2;        // 0..63: row this thread stages
  const int lc = (t & 3) << 2;  // 0,4,8,12: 4 consecutive floats (one b128)

  const uint32_t ldsA = (uint32_t)(uintptr_t)&lds[lr * ASTR + lc];
  const uint32_t ldsB = ldsA + TILE_F * 4;  // bytes
  const float* ga = A + (size_t)lr * lda + lc;
  const float* gb = B + (size_t)lr * ldb + lc;

  // prologue: stage chunk 0 into buffer 0
  async_cp16(ga, ldsA);
  async_cp16(gb, ldsB);

  const int nchunk = kdim / BK;
  for (int ch = 0; ch < nchunk; ++ch) {
    if (ch + 1 < nchunk) {
      // prefetch next chunk into the other buffer
      const uint32_t boff = (uint32_t)(((ch + 1) & 1) * (BUF_F * 4));
      async_cp16(ga + (size_t)(ch + 1) * BK, ldsA + boff);
      async_cp16(gb + (size_t)(ch + 1) * BK, ldsB + boff);
      WAIT_ASYNC(0x2);  // in-order completion: current chunk has landed
    } else {
      WAIT_ASYNC(0x0);
    }
    __syncthreads();
    const float* As = lds + (ch & 1) * BUF_F;
    const float* Bs = As + TILE_F;
    const float* Aw = As + (16 * wr + ml) * ASTR;
    const float* B0 = Bs + (32 * wc + ml) * ASTR;
    const float* B1 = B0 + 16 * ASTR;
#pragma unroll
    for (int s = 0; s < 4; ++s) {
      const int k0 = 4 * s + ksl;
      v2f a, b0, b1;
      a.x  = Aw[k0]; a.y  = Aw[k0 + 1];
      b0.x = B0[k0]; b0.y = B0[k0 + 1];
      b1.x = B1[k0]; b1.y = B1[k0 + 1];
      acc[0] = wmma_f32(a, b0, acc[0]);
      acc[1] = wmma_f32(a, b1, acc[1]);
    }
    __syncthreads();  // protects buffer reuse by the async prefetch
  }
}

// C = X @ W^T + bias.  X:[4096,1024], W:[1024,1024], bias:[1024].
// head_split=1: write C[b,h,s,dk] layout; else flat [row,1024].
__global__ __launch_bounds__(256) void gemm1024_kernel(
    const float* __restrict__ X, const float* __restrict__ W,
    const float* __restrict__ bias, float* __restrict__ C, int head_split) {
  __shared__ float lds[2 * BUF_F];
  const int nT = blockIdx.x * BN;
  const int mT = blockIdx.y * BM;
  v8f acc[2] = {};
  gemm_nt_tile(X + (size_t)mT * DMODEL, DMODEL,
               W + (size_t)nT * DMODEL, DMODEL, DMODEL, lds, acc);
  const int t = threadIdx.x, lane = t & 31, w = t >> 5;
  const int wr = w & 3, wc = w >> 2, ml = lane & 15;
  const int rbase = mT + 16 * wr + ((lane >> 4) << 3);
#pragma unroll
  for (int j = 0; j < 2; ++j) {
    const int col = nT + 32 * wc + 16 * j + ml;
    const float bv = bias[col];
#pragma unroll
    for (int i = 0; i < 8; ++i) {
      const int row = rbase + i;
      const float val = acc[j][i] + bv;
      size_t idx;
      if (head_split) {
        const int b = row >> 11, s = row & (SEQ - 1);
        const int h = col >> 6, dk = col & (DKH - 1);
        idx = ((size_t)((b * NHEADS + h) * SEQ + s)) * DKH + dk;
      } else {
        idx = (size_t)row * DMODEL + col;
      }
      C[idx] = val;
    }
  }
}

// Raw scores per (b,h): attn[bh,q,k] = 0.125 * Q[bh,q,:] . K[bh,k,:]
__global__ __launch_bounds__(256) void scores_kernel(
    const float* __restrict__ Qh, const float* __restrict__ Kh,
    float* __restrict__ attn) {
  __shared__ float lds[2 * BUF_F];
  const int nT = blockIdx.x * BN;  // key block
  const int mT = blockIdx.y * BM;  // query block
  const int bh = blockIdx.z;
  const float* Q = Qh + (size_t)bh * SEQ * DKH;
  const float* K = Kh + (size_t)bh * SEQ * DKH;
  float* outp = attn + (size_t)bh * SEQ * SEQ;
  v8f acc[2] = {};
  gemm_nt_tile(Q + (size_t)mT * DKH, DKH, K + (size_t)nT * DKH, DKH, DKH, lds, acc);
  const int t = threadIdx.x, lane = t & 31, w = t >> 5;
  const int wr = w & 3, wc = w >> 2, ml = lane & 15;
  const int rbase = mT + 16 * wr + ((lane >> 4) << 3);
#pragma unroll
  for (int j = 0; j < 2; ++j) {
    const int col = nT + 32 * wc + 16 * j + ml;
#pragma unroll
    for (int i = 0; i < 8; ++i)
      outp[(size_t)(rbase + i) * SEQ + col] = acc[j][i] * 0.125f;
  }
}

// Per-row softmax stats: stats[row] = {max, 1/sum(exp(x-max))}. One block/row.
__global__ __launch_bounds__(256) void softmax_stats_kernel(
    const float* __restrict__ attn, float* __restrict__ stats) {
  const size_t row = blockIdx.x;
  const float* p = attn + row * SEQ;
  const int t = threadIdx.x, lane = t & 31, w = t >> 5;
  __shared__ float red[8];
  float x[8];
  float m = -3.402823466e38f;
#pragma unroll
  for (int i = 0; i < 8; ++i) { x[i] = p[t + 256 * i]; m = fmaxf(m, x[i]); }
#pragma unroll
  for (int off = 16; off; off >>= 1) m = fmaxf(m, __shfl_xor(m, off, 32));
  if (lane == 0) red[w] = m;
  __syncthreads();
#pragma unroll
  for (int i = 0; i < 8; ++i) m = fmaxf(m, red[i]);
  __syncthreads();
  float s = 0.f;
#pragma unroll
  for (int i = 0; i < 8; ++i) s += __expf(x[i] - m);
#pragma unroll
  for (int off = 16; off; off >>= 1) s += __shfl_xor(s, off, 32);
  if (lane == 0) red[w] = s;
  __syncthreads();
  if (t == 0) {
    float tot = 0.f;
#pragma unroll
    for (int i = 0; i < 8; ++i) tot += red[i];
    stats[row * 2 + 0] = m;
    stats[row * 2 + 1] = 1.0f / tot;
  }
}

// Normalize P in place (writes final attn_weights) and compute AO = P @ V.
// AO written in [B,S,H*64] layout, ready for the output projection.
__global__ __launch_bounds__(256) void attn_pv_kernel(
    float* __restrict__ attn, const float* __restrict__ Vh,
    const float* __restrict__ stats, float* __restrict__ AO) {
  __shared__ float Ps[64 * PSTR];
  __shared__ float Vs[64 * VSTR];
  const int qT = blockIdx.x * 64;
  const int bh = blockIdx.y;
  float* P = attn + (size_t)bh * SEQ * SEQ;
  const float* V = Vh + (size_t)bh * SEQ * DKH;
  const float* st = stats + (size_t)bh * SEQ * 2;
  const int t = threadIdx.x, lane = t & 31, w = t >> 5;
  const int wr = w & 3, wc = w >> 2, ml = lane & 15;
  const int ksl = (lane >> 4) << 1;
  const int lr = t >> 2;         // 0..63
  const int lc = (t & 3) << 4;   // 16 consecutive floats per thread
  const float rmax = st[(qT + lr) * 2 + 0];
  const float rinv = st[(qT + lr) * 2 + 1];
  const uint32_t vlds = (uint32_t)(uintptr_t)&Vs[lr * VSTR + lc];
  v8f acc[2] = {};
  for (int kc = 0; kc < SEQ; kc += 64) {
    // async-stage V tile while we normalize P on the VALU path
    const float* vg = V + (size_t)(kc + lr) * DKH + lc;
#pragma unroll
    for (int q4 = 0; q4 < 4; ++q4)
      async_cp16(vg + 4 * q4, vlds + 16 * q4);
    float* pg = P + (size_t)(qT + lr) * SEQ + kc + lc;
#pragma unroll
    for (int jj = 0; jj < 16; ++jj) {
      const float pv = __expf(pg[jj] - rmax) * rinv;
      pg[jj] = pv;                       // final normalized attn weight
      Ps[lr * PSTR + lc + jj] = pv;
    }
    WAIT_ASYNC(0x0);
    __syncthreads();
    const float* Pw = Ps + (16 * wr + ml) * PSTR;
#pragma unroll
    for (int s = 0; s < 16; ++s) {
      const int k0 = 4 * s + ksl;
      v2f a, b0, b1;
      a.x = Pw[k0];  a.y = Pw[k0 + 1];
      b0.x = Vs[(size_t)k0 * VSTR + 32 * wc + ml];
      b0.y = Vs[(size_t)(k0 + 1) * VSTR + 32 * wc + ml];
      b1.x = Vs[(size_t)k0 * VSTR + 32 * wc + 16 + ml];
      b1.y = Vs[(size_t)(k0 + 1) * VSTR + 32 * wc + 16 + ml];
      acc[0] = wmma_f32(a, b0, acc[0]);
      acc[1] = wmma_f32(a, b1, acc[1]);
    }
    __syncthreads();
  }
  const int b = bh >> 4, h = bh & 15;
  const int rb = qT + 16 * wr + ((lane >> 4) << 3);
#pragma unroll
  for (int j = 0; j < 2; ++j) {
    const int dk = 32 * wc + 16 * j + ml;
#pragma unroll
    for (int i = 0; i < 8; ++i) {
      const int q = rb + i;
      AO[((size_t)(b * SEQ + q)) * DMODEL + h * DKH + dk] = acc[j][i];
    }
  }
}

extern "C" void kernel_launch(void* const* d_in, const int* in_sizes, int n_in,
                              void* d_out, int out_size, void* d_ws, size_t ws_size,
                              hipStream_t stream) {
  const float* query = (const float*)d_in[0];
  const float* key   = (const float*)d_in[1];
  const float* value = (const float*)d_in[2];
  const float* Wq = (const float*)d_in[3];
  const float* bq = (const float*)d_in[4];
  const float* Wk = (const float*)d_in[5];
  const float* bk = (const float*)d_in[6];
  const float* Wv = (const float*)d_in[7];
  const float* bv = (const float*)d_in[8];
  const float* Wo = (const float*)d_in[9];
  const float* bo = (const float*)d_in[10];

  // d_out = [output | attn_weights | K | V], flat fp32.
  float* out  = (float*)d_out;
  float* attn = out + (size_t)NBATCH * SEQ * DMODEL;
  float* Kout = attn + (size_t)NBATCH * NHEADS * SEQ * SEQ;
  float* Vout = Kout + (size_t)NBATCH * NHEADS * SEQ * DKH;

  // workspace: Qh (16 MB) | AO (16 MB) | softmax stats (512 KB)
  float* Qh    = (float*)d_ws;
  float* AO    = Qh + (size_t)NBATCH * NHEADS * SEQ * DKH;
  float* stats = AO + (size_t)NBATCH * SEQ * DMODEL;

  dim3 gproj(DMODEL / BN, (NBATCH * SEQ) / BM);      // (16, 64)
  gemm1024_kernel<<<gproj, 256, 0, stream>>>(query, Wq, bq, Qh, 1);
  gemm1024_kernel<<<gproj, 256, 0, stream>>>(key,   Wk, bk, Kout, 1);
  gemm1024_kernel<<<gproj, 256, 0, stream>>>(value, Wv, bv, Vout, 1);

  dim3 gsc(SEQ / BN, SEQ / BM, NBATCH * NHEADS);     // (32, 32, 32)
  scores_kernel<<<gsc, 256, 0, stream>>>(Qh, Kout, attn);

  softmax_stats_kernel<<<NBATCH * NHEADS * SEQ, 256, 0, stream>>>(attn, stats);

  dim3 gpv(SEQ / 64, NBATCH * NHEADS);               // (32, 32)
  attn_pv_kernel<<<gpv, 256, 0, stream>>>(attn, Vout, stats, AO);

  gemm1024_kernel<<<gproj, 256, 0, stream>>>(AO, Wo, bo, out, 0);
}